// Inv1d_79791902425207
// MI455X (gfx1250) — compile-verified
//
#include <hip/hip_runtime.h>
#include <hip/hip_bf16.h>

// ---------------------------------------------------------------------------
// Fused involution-1d for MI455X (gfx1250, wave32).
//   CH=256, G=16, GROUP=16, K=7, PAD=3, STRIDE=1 (pool == identity)
//   b=16, dim=4096, hid=64
// One workgroup = one (batch, 64-wide L tile). x tile staged into LDS via
// CDNA5 async global->LDS copies (ASYNCcnt), generator GEMMs done with
// V_WMMA_F32_16X16X4_F32, involution epilogue reads LDS and writes out
// coalesced. Memory-bound: x read once, out written once (~128 MB total ->
// ~5.5 us at 23.3 TB/s HBM).
// ---------------------------------------------------------------------------

#define CH    256
#define DIM   4096
#define HID   64
#define GK    112        // K*G
#define LT    64         // L tile
#define PADK  3
#define XCOLS 72         // LT + 8 halo (aligned for b128)
#define XSTR  72
#define XOFF  4          // column 4 of the tile == global l0
#define HSTR  64
#define KSTR  64

typedef float v2f __attribute__((ext_vector_type(2)));
typedef float v8f __attribute__((ext_vector_type(8)));

// smem layout (floats):
//   xs : 256*72 = 18432   (starts at dynamic-LDS offset 0)
//   hs : 64*64  =  4096
//   ks : 112*64 =  7168
//   s_b1/s_bnA/s_bnB : 64 each ; s_b2 : 112
#define SMEM_FLOATS (CH*XSTR + HID*HSTR + GK*KSTR + 3*HID + GK)

__global__ __launch_bounds__(256) void inv1d_fused(
    const float* __restrict__ x,
    const float* __restrict__ w1, const float* __restrict__ b1,
    const float* __restrict__ bn_gamma, const float* __restrict__ bn_beta,
    const float* __restrict__ bn_mean,  const float* __restrict__ bn_var,
    const float* __restrict__ w2, const float* __restrict__ b2,
    float* __restrict__ out)
{
    extern __shared__ float smem[];
    float* xs   = smem;
    float* hs   = xs + CH * XSTR;
    float* ks   = hs + HID * HSTR;
    float* s_b1 = ks + GK * KSTR;
    float* s_bnA = s_b1 + HID;
    float* s_bnB = s_bnA + HID;
    float* s_b2  = s_bnB + HID;

    const int tid  = threadIdx.x;
    const int wave = tid >> 5;
    const int lane = tid & 31;
    const int lhalf = lane >> 4;       // 0: lanes 0-15, 1: lanes 16-31
    const int lmod  = lane & 15;
    const int koff  = lhalf << 1;      // A/B frag K offset (v0->K0/K2, v1->K1/K3)

    const int tile  = blockIdx.x;      // 0..63
    const int batch = blockIdx.y;      // 0..15
    const int l0    = tile * LT;
    const float* xb = x + (size_t)batch * CH * DIM;

    // ---- stage small params ----
    if (tid < HID) {
        float a = bn_gamma[tid] * rsqrtf(bn_var[tid] + 1e-5f);
        s_bnA[tid] = a;
        s_bnB[tid] = bn_beta[tid] - bn_mean[tid] * a;
        s_b1[tid]  = b1[tid];
    }
    if (tid < GK) s_b2[tid] = b2[tid];

    // ---- stage x tile [256][72] with halo (cols map to global l0-4+j) ----
    const int gl0 = l0 - XOFF;
    if (gl0 >= 0 && gl0 + XCOLS <= DIM) {
        // Interior tile: async global->LDS b128 copies (no VGPR round-trip,
        // tracked by ASYNCcnt). Dynamic LDS starts at offset 0, so the LDS
        // address operand is the byte offset of xs[...] inside smem.
        for (int i = tid; i < CH * (XCOLS / 4); i += 256) {
            int c = i / (XCOLS / 4);
            int j = (i - c * (XCOLS / 4)) * 4;
            unsigned int ldsoff = (unsigned int)((c * XSTR + j) * sizeof(float));
            unsigned long long gaddr =
                (unsigned long long)(uintptr_t)(xb + (size_t)c * DIM + gl0 + j);
            asm volatile("global_load_async_to_lds_b128 %0, %1, off"
                         :: "v"(ldsoff), "v"(gaddr)
                         : "memory");
        }
        asm volatile("s_wait_asynccnt 0x0" ::: "memory");
    } else {
        for (int i = tid; i < CH * XCOLS; i += 256) {
            int c = i / XCOLS;
            int j = i - c * XCOLS;
            int gl = gl0 + j;
            xs[c * XSTR + j] = (gl >= 0 && gl < DIM) ? xb[(size_t)c * DIM + gl] : 0.0f;
        }
    }

    // prefetch next tile of x into L2 while we compute (global_prefetch_b8)
    if (l0 + LT < DIM)
        __builtin_prefetch(xb + (size_t)tid * DIM + l0 + LT, 0, 0);

    __syncthreads();

    // ---- GEMM1: h[64][64] = w1[64][256] @ xt[256][64], fused bias/relu/BN ----
    // 16 C-tiles (4x4 of 16x16), round-robin over 8 waves.
    for (int t = wave; t < 16; t += 8) {
        const int m0 = (t >> 2) << 4;
        const int n0 = (t & 3) << 4;
        v8f acc = {};
        for (int k = 0; k < CH; k += 4) {
            v2f afrag, bfrag;
            const float* w1p = w1 + (m0 + lmod) * CH + k + koff;
            afrag.x = w1p[0];
            afrag.y = w1p[1];
            const float* xp = &xs[(k + koff) * XSTR + XOFF + n0 + lmod];
            bfrag.x = xp[0];
            bfrag.y = xp[XSTR];
            acc = __builtin_amdgcn_wmma_f32_16x16x4_f32(
                false, afrag, false, bfrag, (short)0, acc, false, false);
        }
        #pragma unroll
        for (int r = 0; r < 8; ++r) {
            int row = m0 + r + (lhalf << 3);
            int col = n0 + lmod;
            float v = acc[r] + s_b1[row];
            v = fmaxf(v, 0.0f);
            hs[row * HSTR + col] = fmaf(s_bnA[row], v, s_bnB[row]);
        }
    }
    __syncthreads();

    // ---- GEMM2: kern[112][64] = w2[112][64] @ h[64][64] + b2 ----
    // 28 C-tiles (7x4 of 16x16), round-robin over 8 waves.
    for (int t = wave; t < 28; t += 8) {
        const int m0 = (t >> 2) << 4;
        const int n0 = (t & 3) << 4;
        v8f acc = {};
        for (int k = 0; k < HID; k += 4) {
            v2f afrag, bfrag;
            const float* w2p = w2 + (m0 + lmod) * HID + k + koff;
            afrag.x = w2p[0];
            afrag.y = w2p[1];
            const float* hp = &hs[(k + koff) * HSTR + n0 + lmod];
            bfrag.x = hp[0];
            bfrag.y = hp[HSTR];
            acc = __builtin_amdgcn_wmma_f32_16x16x4_f32(
                false, afrag, false, bfrag, (short)0, acc, false, false);
        }
        #pragma unroll
        for (int r = 0; r < 8; ++r) {
            int row = m0 + r + (lhalf << 3);
            int col = n0 + lmod;
            ks[row * KSTR + col] = acc[r] + s_b2[row];
        }
    }
    __syncthreads();

    // ---- involution: out[c][l] = sum_k ks[(c/16)*7+k][l] * xs[c][l+k+1] ----
    // thread mapping: lidx = tid&63 -> coalesced 64-wide stores per c
    const int lidx = tid & 63;
    const int cq   = tid >> 6;       // 0..3
    float* ob = out + (size_t)batch * CH * DIM + l0 + lidx;
    for (int ci = 0; ci < 64; ++ci) {
        const int c = (cq << 6) + ci;
        const int g = c >> 4;
        const float* kr = &ks[(g * 7) * KSTR + lidx];
        const float* xr = &xs[c * XSTR + lidx + (XOFF - PADK)];
        float acc = 0.0f;
        #pragma unroll
        for (int k = 0; k < 7; ++k)
            acc = fmaf(kr[k * KSTR], xr[k], acc);
        ob[(size_t)c * DIM] = acc;
    }
}

extern "C" void kernel_launch(void* const* d_in, const int* in_sizes, int n_in,
                              void* d_out, int out_size, void* d_ws, size_t ws_size,
                              hipStream_t stream) {
    (void)in_sizes; (void)n_in; (void)d_ws; (void)ws_size; (void)out_size;
    const float* x        = (const float*)d_in[0];
    const float* w1       = (const float*)d_in[1];
    const float* b1       = (const float*)d_in[2];
    const float* bn_gamma = (const float*)d_in[3];
    const float* bn_beta  = (const float*)d_in[4];
    const float* bn_mean  = (const float*)d_in[5];
    const float* bn_var   = (const float*)d_in[6];
    const float* w2       = (const float*)d_in[7];
    const float* b2       = (const float*)d_in[8];
    float* out = (float*)d_out;

    const size_t smem_bytes = SMEM_FLOATS * sizeof(float);   // ~117 KB of the 320 KB WGP LDS
    (void)hipFuncSetAttribute((const void*)inv1d_fused,
                              hipFuncAttributeMaxDynamicSharedMemorySize,
                              (int)smem_bytes);

    dim3 grid(DIM / LT, 16);   // 64 L-tiles x 16 batches = 1024 WGs
    dim3 block(256);           // 8 wave32
    inv1d_fused<<<grid, block, smem_bytes, stream>>>(
        x, w1, b1, bn_gamma, bn_beta, bn_mean, bn_var, w2, b2, out);
}